// SPDRectified_20959440405162
// MI455X (gfx1250) — compile-verified
//
#include <hip/hip_runtime.h>

// SPD rectification via Newton–Schulz matrix-sign iteration (GEMM-only, no eigh):
//   out = 0.5*(X + eps*I + abs(X - eps*I)),  abs(B) = sign(B)*B
//   sign via S_{k+1} = S_k*(1.5*I - 0.5*S_k*S_k), S_0 = B/||B||_F
// All GEMMs are 64x64x64 fp32 in LDS using V_WMMA_F32_16X16X4_F32 (CDNA5, wave32).
// The affine transform (1.5*I - 0.5*Y) is fused into the first GEMM's accumulator
// writeback, eliminating the elementwise pass and one barrier per iteration.

typedef __attribute__((ext_vector_type(2))) float v2f;
typedef __attribute__((ext_vector_type(8))) float v8f;

#define MDIM     64
#define LSTR     68           // padded LDS row stride (floats) to spread banks
#define NS_ITERS 16
#define EPSV     1e-4f

// Workgroup-cooperative 64x64x64 GEMM in LDS:
//   FIX=false: C = A*B
//   FIX=true : C = 1.5*I - 0.5*(A*B)        (fused Newton-Schulz polynomial step)
// 8 waves; wave w owns output tiles (ti0, tj) and (ti0+2, tj) with
//   ti0 = w>>2, tj = w&3  -> both tiles share the SAME B fragments.
// One fused K-loop: load b once, two independent WMMA accumulator chains.
// EXEC is all-ones (uniform control flow) as WMMA requires.
// Fragment layouts per CDNA5 ISA 7.12.2 (f32, wave32):
//   A 16x4 : lane L (M = L&15, half = L>>4): v2f = { A[M][k0+2*half], A[M][k0+2*half+1] }
//   B 4x16 : lane L (N = L&15, half = L>>4): v2f = { B[k0+2*half][N], B[k0+2*half+1][N] }
//   C 16x16: VGPR v, lane L: C[v + 8*(L>>4)][L&15]
template <bool FIX>
__device__ __forceinline__ void gemm64_wmma(const float* __restrict__ A,
                                            const float* __restrict__ B,
                                            float* __restrict__ C,
                                            int wave, int lane) {
    const int M    = lane & 15;
    const int half = lane >> 4;
    const int ti0  = wave >> 2;          // 0..1
    const int ti1  = ti0 + 2;            // 2..3
    const int tj   = wave & 3;

    v8f c0 = {0.f, 0.f, 0.f, 0.f, 0.f, 0.f, 0.f, 0.f};
    v8f c1 = {0.f, 0.f, 0.f, 0.f, 0.f, 0.f, 0.f, 0.f};

    // 8-byte aligned: LSTR even, 2*half even, k0 multiple of 4.
    const float* A0 = A + (ti0 * 16 + M) * LSTR + 2 * half;
    const float* A1 = A + (ti1 * 16 + M) * LSTR + 2 * half;
    const float* Bc = B + (2 * half) * LSTR + tj * 16 + M;

#pragma unroll
    for (int k0 = 0; k0 < MDIM; k0 += 4) {
        v2f a0 = *(const v2f*)(A0 + k0);          // merged into ds_load_2addr_b64
        v2f a1 = *(const v2f*)(A1 + k0);
        v2f b;
        b.x = Bc[k0 * LSTR];                       // strided pair, loaded ONCE
        b.y = Bc[k0 * LSTR + LSTR];
        c0 = __builtin_amdgcn_wmma_f32_16x16x4_f32(false, a0, false, b,
                                                   (short)0, c0, false, false);
        c1 = __builtin_amdgcn_wmma_f32_16x16x4_f32(false, a1, false, b,
                                                   (short)0, c1, false, false);
    }

    const int n = tj * 16 + M;
    float* Crow = C + n;
#pragma unroll
    for (int v = 0; v < 8; ++v) {
        const int m0 = ti0 * 16 + v + 8 * half;
        const int m1 = ti1 * 16 + v + 8 * half;
        float w0 = c0[v];
        float w1 = c1[v];
        if (FIX) {
            w0 = -0.5f * w0 + ((m0 == n) ? 1.5f : 0.f);
            w1 = -0.5f * w1 + ((m1 == n) ? 1.5f : 0.f);
        }
        Crow[m0 * LSTR] = w0;
        Crow[m1 * LSTR] = w1;
    }
}

__global__ void __launch_bounds__(256)
spd_rectify_ns_kernel(const float* __restrict__ x, float* __restrict__ out) {
    extern __shared__ float lds[];
    float* buf0  = lds;                      // S
    float* buf1  = lds + 1 * MDIM * LSTR;    // S_next
    float* buf2  = lds + 2 * MDIM * LSTR;    // T (poly scratch / B staging)
    float* accum = lds + 3 * MDIM * LSTR;    // Frobenius accumulator

    const int tid  = threadIdx.x;
    const int wave = tid >> 5;
    const int lane = tid & 31;
    const int row  = tid >> 2;               // 0..63 (4 threads per row)
    const int col0 = (tid & 3) * 16;         // quarter-row start
    const long base = (long)blockIdx.x * (MDIM * MDIM);

    // ---- Load quarter-row of X, form B = X - eps*I in registers ----
    float breg[16];
    {
        const float4* src = (const float4*)(x + base + row * MDIM + col0);
        float4 bv[4];
#pragma unroll
        for (int i = 0; i < 4; ++i) bv[i] = src[i];
#pragma unroll
        for (int i = 0; i < 4; ++i) {
            breg[4 * i + 0] = bv[i].x;
            breg[4 * i + 1] = bv[i].y;
            breg[4 * i + 2] = bv[i].z;
            breg[4 * i + 3] = bv[i].w;
        }
        if (row >= col0 && row < col0 + 16) breg[row - col0] -= EPSV;
    }

    // ---- Frobenius norm of B (ds_add_f32 reduction) ----
    if (tid == 0) *accum = 0.f;
    __syncthreads();
    {
        float psum = 0.f;
#pragma unroll
        for (int j = 0; j < 16; ++j) psum += breg[j] * breg[j];
        atomicAdd(accum, psum);
    }
    __syncthreads();
    const float nf2  = *accum;
    const float invn = (nf2 > 1e-30f) ? rsqrtf(nf2) : 0.f;   // 1/||B||_F

    // ---- Seed S0 = B / ||B||_F into LDS ----
#pragma unroll
    for (int j = 0; j < 16; ++j) buf0[row * LSTR + col0 + j] = breg[j] * invn;
    __syncthreads();

    // ---- Newton–Schulz sign iteration: 2 GEMMs + 2 barriers per iter ----
    float* S  = buf0;
    float* Sn = buf1;
    float* T  = buf2;
    for (int it = 0; it < NS_ITERS; ++it) {
        gemm64_wmma<true >(S, S, T,  wave, lane);  // T  = 1.5*I - 0.5*S*S (fused)
        __syncthreads();
        gemm64_wmma<false>(S, T, Sn, wave, lane);  // Sn = S*T
        __syncthreads();
        float* tmp = S; S = Sn; Sn = tmp;          // uniform swap
    }

    // ---- W = sign(B) * B ----
#pragma unroll
    for (int j = 0; j < 16; ++j) T[row * LSTR + col0 + j] = breg[j];
    __syncthreads();
    gemm64_wmma<false>(S, T, Sn, wave, lane);      // Sn = S * B = abs(B)
    __syncthreads();

    // ---- out = 0.5*(B + W) + eps*I  (== 0.5*(X + eps*I + abs(X-eps*I))) ----
    {
        float4 ov[4];
#pragma unroll
        for (int i = 0; i < 4; ++i) {
            float o[4];
#pragma unroll
            for (int q = 0; q < 4; ++q) {
                const int j = 4 * i + q;
                float v = 0.5f * (breg[j] + Sn[row * LSTR + col0 + j]);
                if (col0 + j == row) v += EPSV;
                o[q] = v;
            }
            ov[i] = make_float4(o[0], o[1], o[2], o[3]);
        }
        float4* dst = (float4*)(out + base + row * MDIM + col0);
#pragma unroll
        for (int i = 0; i < 4; ++i) dst[i] = ov[i];
    }
}

extern "C" void kernel_launch(void* const* d_in, const int* in_sizes, int n_in,
                              void* d_out, int out_size, void* d_ws, size_t ws_size,
                              hipStream_t stream) {
    const float* x   = (const float*)d_in[0];
    float*       out = (float*)d_out;
    const int batch  = in_sizes[0] / (MDIM * MDIM);   // 16384
    const size_t shmem = (size_t)(3 * MDIM * LSTR + 1) * sizeof(float);  // ~52 KB
    spd_rectify_ns_kernel<<<batch, 256, shmem, stream>>>(x, out);
}